// UnitaryBranching_88716844466896
// MI455X (gfx1250) — compile-verified
//
#include <hip/hip_runtime.h>

typedef __attribute__((ext_vector_type(2))) float v2f;
typedef __attribute__((ext_vector_type(8))) float v8f;

#define DIM 64
#define NPOW 17          // P^0 .. P^16
#define MAT_ELEMS (DIM * DIM)

// ---------------------------------------------------------------------------
// Kernel 1: build P^0..P^16 where P = primitives[1].
// The reference's clamped step table only ever applies primitives[1] or the
// identity, so every output map is a pure power of primitives[1]; building the
// powers left-to-right (M <- M*P) reproduces the reference's fp op sequence
// exactly (identity multiplies are bit-preserving).
//
// One workgroup, 16 wave32s; wave w owns output tile (w/4, w%4) of the 64x64
// product and sweeps K=64 with sixteen v_wmma_f32_16x16x4_f32 ops from LDS.
// ---------------------------------------------------------------------------
__global__ void __launch_bounds__(512)
pow_chain_kernel(const float* __restrict__ primitives, float* __restrict__ pows) {
    __shared__ float Msh[MAT_ELEMS];   // running product M = P^(p-1)
    __shared__ float Psh[MAT_ELEMS];   // P = primitives[1]

    const int tid  = threadIdx.x;
    const int lane = tid & 31;
    const int wave = tid >> 5;     // 0..15
    const int ti   = wave >> 2;    // tile row 0..3
    const int tj   = wave & 3;     // tile col 0..3

    // Load P = primitives[1]; init M = I; emit P^0 = I.
    for (int e = tid; e < MAT_ELEMS; e += 512) {
        Psh[e] = primitives[MAT_ELEMS + e];     // primitives[1]
        const float idv = ((e >> 6) == (e & 63)) ? 1.0f : 0.0f;
        Msh[e]  = idv;
        pows[e] = idv;
    }
    __syncthreads();

    // WMMA 16x16x4 f32 fragment indexing (ISA 7.12.2):
    //  A 16x4:  lanes 0-15 hold K=0,1; lanes 16-31 hold K=2,3 (per-row float2)
    //  B 4x16:  VGPR0/1 = rows {0,1} (lanes 0-15) / rows {2,3} (lanes 16-31)
    //  C/D 16x16: VGPR r = row r (lanes 0-15) / row r+8 (lanes 16-31)
    const int m     = lane & 15;   // A row / B-C-D column within tile
    const int khalf = lane >> 4;   // which K pair this half-wave covers

    for (int p = 1; p < NPOW; ++p) {
        v8f acc = {};
        #pragma unroll
        for (int kb = 0; kb < DIM; kb += 4) {
            const int k0 = kb + 2 * khalf;
            v2f a, b;
            a.x = Msh[(ti * 16 + m) * DIM + k0 + 0];
            a.y = Msh[(ti * 16 + m) * DIM + k0 + 1];
            b.x = Psh[(k0 + 0) * DIM + tj * 16 + m];
            b.y = Psh[(k0 + 1) * DIM + tj * 16 + m];
            // 8 args: (neg_a, A, neg_b, B, c_mod, C, reuse_a, reuse_b)
            acc = __builtin_amdgcn_wmma_f32_16x16x4_f32(
                false, a, false, b, (short)0, acc, false, false);
        }
        __syncthreads();   // every wave done reading Msh before overwrite

        float* __restrict__ dst = pows + (size_t)p * MAT_ELEMS;
        #pragma unroll
        for (int r = 0; r < 8; ++r) {
            const int row = ti * 16 + r + 8 * khalf;
            const int col = tj * 16 + m;
            Msh[row * DIM + col] = acc[r];
            dst[row * DIM + col] = acc[r];
        }
        __syncthreads();
    }
}

// ---------------------------------------------------------------------------
// Kernel 2: per position, derive the power count c from the tree index
// (faithful peel: idx -> (idx-2)/2+1, step==1 <=> (idx-2)%2==0) and stream
// P^c to the output with 128-bit coalesced loads/stores. 17 matrices = 272 KB
// stay L2-resident; the 256 MB of stores are the roofline (~11 us @ 23.3 TB/s).
// ---------------------------------------------------------------------------
__global__ void __launch_bounds__(256)
gather_pow_kernel(const int* __restrict__ unique,
                  const float* __restrict__ pows,
                  float* __restrict__ out) {
    const int n = blockIdx.x;

    int idx = unique[n];
    int c = 0;
    #pragma unroll
    for (int d = 0; d < 16; ++d) {
        if (idx > 1) {
            c += (((idx - 2) & 1) == 0) ? 1 : 0;
            idx = ((idx - 2) >> 1) + 1;
        }
    }

    const float4* __restrict__ src = (const float4*)(pows + (size_t)c * MAT_ELEMS);
    float4* __restrict__ dst       = (float4*)(out + (size_t)n * MAT_ELEMS);
    #pragma unroll
    for (int e = threadIdx.x; e < MAT_ELEMS / 4; e += 256)
        dst[e] = src[e];
}

// ---------------------------------------------------------------------------
// inputs (setup_inputs order): unique[int32 16384], primitives[f32 2*64*64],
// identity[f32 1*64*64]. output: f32 16384*64*64. ws: needs 17*64*64*4 =
// 278,528 bytes for the power table.
// ---------------------------------------------------------------------------
extern "C" void kernel_launch(void* const* d_in, const int* in_sizes, int n_in,
                              void* d_out, int out_size, void* d_ws, size_t ws_size,
                              hipStream_t stream) {
    const int*   unique     = (const int*)d_in[0];
    const float* primitives = (const float*)d_in[1];
    (void)d_in; (void)n_in; (void)ws_size; (void)out_size;

    float* pows = (float*)d_ws;
    float* out  = (float*)d_out;
    const int n = in_sizes[0];   // 16384 positions

    pow_chain_kernel<<<dim3(1), dim3(512), 0, stream>>>(primitives, pows);
    gather_pow_kernel<<<dim3(n), dim3(256), 0, stream>>>(unique, pows, out);
}